// IntraGraphAttention_4501125726636
// MI455X (gfx1250) — compile-verified
//
#include <hip/hip_runtime.h>
#include <math.h>

typedef float v2f __attribute__((ext_vector_type(2)));
typedef float v8f __attribute__((ext_vector_type(8)));

#define D_IN 128
#define HC   64   // H*C = 2*32

// ---------- helpers ----------
__device__ __forceinline__ unsigned f2mono(float f) {
    unsigned u = __float_as_uint(f);
    return (u & 0x80000000u) ? ~u : (u | 0x80000000u);
}
__device__ __forceinline__ float mono2f(unsigned u) {
    return (u & 0x80000000u) ? __uint_as_float(u & 0x7fffffffu)
                             : __uint_as_float(~u);
}
__device__ __forceinline__ void edge_sd(const int* __restrict__ ei, int E, int e,
                                        int& s, int& d) {
    if (e < E) { s = ei[e]; d = ei[E + e]; }
    else       { s = d = e - E; }           // self loop
}
__device__ __forceinline__ float lrelu02(float v) { return v > 0.f ? v : 0.2f * v; }

// ---------- K0: zero-init output + scratch ----------
__global__ void k_init(float* __restrict__ out, unsigned* __restrict__ mEnc,
                       float* __restrict__ denom, float* __restrict__ nwsum,
                       float* __restrict__ cnt, int n) {
    int t = blockIdx.x * blockDim.x + threadIdx.x;
    if (t < n * HC) out[t] = 0.f;
    if (t < n * 2) { mEnc[t] = 0u; denom[t] = 0.f; }   // mono(x) > 0 for all finite x
    if (t < n)     { nwsum[t] = 0.f; cnt[t] = 0.f; }
}

// ---------- K1: h = elu(x) @ W via V_WMMA_F32_16X16X4_F32 ----------
// block = 128 threads = 4 waves; block computes rows [16b,16b+16), wave w -> cols [16w,16w+16)
__global__ __launch_bounds__(128)
void k_gemm_wmma(const float* __restrict__ x, const float* __restrict__ W,
                 float* __restrict__ h, int n) {
    __shared__ float As[16][D_IN + 4];   // +4 pad: stride 132 -> conflict-light
    const int lane = threadIdx.x & 31;
    const int wave = threadIdx.x >> 5;
    const int row0 = blockIdx.x * 16;

    // cooperative elu(x) tile load: thread t -> row t>>3, 16 cols at (t&7)*16
    {
        int r  = threadIdx.x >> 3;
        int c0 = (threadIdx.x & 7) * 16;
        int gr = row0 + r; if (gr >= n) gr = n - 1;     // clamp (keeps EXEC full later)
        const float* xp = x + (size_t)gr * D_IN + c0;
#pragma unroll
        for (int i = 0; i < 16; ++i) {
            float v = xp[i];
            As[r][c0 + i] = v > 0.f ? v : (__expf(v) - 1.f);
        }
    }
    __syncthreads();

    const int col0    = wave * 16;
    const int halfSel = lane >> 4;     // 0: lanes 0-15, 1: lanes 16-31
    const int lrow    = lane & 15;
    v8f acc = {};
#pragma unroll 4
    for (int k = 0; k < D_IN; k += 4) {
        // A 16x4 frag: lanes0-15 hold K=k..k+1, lanes16-31 hold K=k+2..k+3 (ISA layout)
        v2f a, b;
        a.x = As[lrow][k + 2 * halfSel + 0];
        a.y = As[lrow][k + 2 * halfSel + 1];
        // B 4x16 frag, N = col0+lrow, same K split across lane halves
        const float* wp = W + (size_t)(k + 2 * halfSel) * HC + col0 + lrow;
        b.x = wp[0];
        b.y = wp[HC];
        acc = __builtin_amdgcn_wmma_f32_16x16x4_f32(false, a, false, b,
                                                    (short)0, acc, false, false);
    }
    // C/D layout: VGPR r -> row r (lanes0-15) / row 8+r (lanes16-31), col = col0 + lane%16
    float* hp = h + (size_t)(row0 + 8 * halfSel) * HC + col0 + lrow;
    if (row0 + 16 <= n) {
        // full tile: branch-free strided stores
#pragma unroll
        for (int r = 0; r < 8; ++r) hp[(size_t)r * HC] = acc[r];
    } else {
#pragma unroll
        for (int r = 0; r < 8; ++r) {
            if (row0 + r + 8 * halfSel < n) hp[(size_t)r * HC] = acc[r];
        }
    }
}

// ---------- K2: a_src/a_dst per (node, head), vectorized b128 reads ----------
__global__ void k_attn(const float* __restrict__ h, const float* __restrict__ att_src,
                       const float* __restrict__ att_dst, float* __restrict__ a_src,
                       float* __restrict__ a_dst, int n) {
    int t = blockIdx.x * blockDim.x + threadIdx.x;
    if (t >= n * 2) return;
    int node = t >> 1, head = t & 1;
    const float4* hp = (const float4*)(h + (size_t)node * HC + head * 32);
    const float4* as = (const float4*)(att_src + head * 32);
    const float4* ad = (const float4*)(att_dst + head * 32);
    float s = 0.f, d = 0.f;
#pragma unroll
    for (int c = 0; c < 8; ++c) {
        float4 v = hp[c], vs = as[c], vd = ad[c];
        s += v.x * vs.x + v.y * vs.y + v.z * vs.z + v.w * vs.w;
        d += v.x * vd.x + v.y * vd.y + v.z * vd.z + v.w * vd.w;
    }
    a_src[t] = s; a_dst[t] = d;
}

// ---------- K3: segment max (monotone-uint atomicMax) ----------
__global__ void k_max(const int* __restrict__ ei, const float* __restrict__ a_src,
                      const float* __restrict__ a_dst, unsigned* __restrict__ mEnc,
                      int E, int n) {
    int t = blockIdx.x * blockDim.x + threadIdx.x;
    if (t >= (E + n) * 2) return;
    int e = t >> 1, head = t & 1;
    int s, d; edge_sd(ei, E, e, s, d);
    float al = lrelu02(a_src[s * 2 + head] + a_dst[d * 2 + head]);
    atomicMax(&mEnc[d * 2 + head], f2mono(al));
}

// ---------- K4: segment sum of exp ----------
__global__ void k_denom(const int* __restrict__ ei, const float* __restrict__ a_src,
                        const float* __restrict__ a_dst, const unsigned* __restrict__ mEnc,
                        float* __restrict__ denom, int E, int n) {
    int t = blockIdx.x * blockDim.x + threadIdx.x;
    if (t >= (E + n) * 2) return;
    int e = t >> 1, head = t & 1;
    int s, d; edge_sd(ei, E, e, s, d);
    float al = lrelu02(a_src[s * 2 + head] + a_dst[d * 2 + head]);
    atomicAdd(&denom[d * 2 + head], __expf(al - mono2f(mEnc[d * 2 + head])));
}

// ---------- K5: weighted aggregate, block = 1 edge, 64 lanes = 64 cols ----------
__global__ __launch_bounds__(64)
void k_agg(const int* __restrict__ ei, const float* __restrict__ a_src,
           const float* __restrict__ a_dst, const unsigned* __restrict__ mEnc,
           const float* __restrict__ denom, const float* __restrict__ h,
           float* __restrict__ out, int E) {
    int e = blockIdx.x;
    int t = threadIdx.x;           // 0..63
    int head = t >> 5;
    int s, d; edge_sd(ei, E, e, s, d);
    float al = lrelu02(a_src[s * 2 + head] + a_dst[d * 2 + head]);
    float p  = __expf(al - mono2f(mEnc[d * 2 + head])) / denom[d * 2 + head];
    atomicAdd(&out[(size_t)d * HC + t], p * h[(size_t)s * HC + t]);
}

// ---------- K6: enrichment scatter-mean accumulation ----------
__global__ void k_enrich(const int* __restrict__ ei, const float* __restrict__ w,
                         float* __restrict__ nwsum, float* __restrict__ cnt, int E) {
    int e = blockIdx.x * blockDim.x + threadIdx.x;
    if (e >= E) return;
    int s = ei[e];
    atomicAdd(&nwsum[s], w[e]);
    atomicAdd(&cnt[s], 1.0f);
}

// ---------- K7: finalize (bias + enrichment term) ----------
__global__ void k_final(float* __restrict__ out, const float* __restrict__ bias,
                        const float* __restrict__ nwsum, const float* __restrict__ cnt,
                        const float* __restrict__ esc, int n) {
    int t = blockIdx.x * blockDim.x + threadIdx.x;
    if (t >= n * HC) return;
    int node = t >> 6, c = t & 63;
    float cc = cnt[node]; cc = cc > 1.f ? cc : 1.f;
    float nw = nwsum[node] / cc;
    nw = fminf(fmaxf(nw, 0.2f), 5.0f);
    float sf = 0.1f / (1.f + __expf(-esc[0]));
    out[t] += bias[c] + sf * (nw - 1.f);
}

extern "C" void kernel_launch(void* const* d_in, const int* in_sizes, int n_in,
                              void* d_out, int out_size, void* d_ws, size_t ws_size,
                              hipStream_t stream) {
    const float* x       = (const float*)d_in[0];
    const int*   ei      = (const int*)  d_in[1];   // [2,E]: row0=src, row1=dst
    const float* iew     = (const float*)d_in[2];
    const float* W       = (const float*)d_in[3];
    const float* att_src = (const float*)d_in[4];
    const float* att_dst = (const float*)d_in[5];
    const float* bias    = (const float*)d_in[6];
    const float* esc     = (const float*)d_in[7];
    float* out = (float*)d_out;

    const int n = in_sizes[0] / D_IN;   // 50000
    const int E = in_sizes[2];          // 1600000

    // workspace layout (all offsets 16B-aligned)
    char* ws = (char*)d_ws;
    float*    h     = (float*)ws;    ws += (size_t)n * HC * sizeof(float);
    float*    a_src = (float*)ws;    ws += (size_t)n * 2  * sizeof(float);
    float*    a_dst = (float*)ws;    ws += (size_t)n * 2  * sizeof(float);
    unsigned* mEnc  = (unsigned*)ws; ws += (size_t)n * 2  * sizeof(unsigned);
    float*    denom = (float*)ws;    ws += (size_t)n * 2  * sizeof(float);
    float*    nwsum = (float*)ws;    ws += (size_t)n      * sizeof(float);
    float*    cnt   = (float*)ws;    /* total n*296 B, ~14.8 MB */

    const int tot   = n * HC;
    const int edges = E + n;            // with self-loops

    k_init  <<<(tot + 255) / 256, 256, 0, stream>>>(out, mEnc, denom, nwsum, cnt, n);
    k_gemm_wmma<<<(n + 15) / 16, 128, 0, stream>>>(x, W, h, n);
    k_attn  <<<(n * 2 + 255) / 256, 256, 0, stream>>>(h, att_src, att_dst, a_src, a_dst, n);
    k_max   <<<(edges * 2 + 255) / 256, 256, 0, stream>>>(ei, a_src, a_dst, mEnc, E, n);
    k_denom <<<(edges * 2 + 255) / 256, 256, 0, stream>>>(ei, a_src, a_dst, mEnc, denom, E, n);
    k_agg   <<<edges, 64, 0, stream>>>(ei, a_src, a_dst, mEnc, denom, h, out, E);
    k_enrich<<<(E + 255) / 256, 256, 0, stream>>>(ei, iew, nwsum, cnt, E);
    k_final <<<(tot + 255) / 256, 256, 0, stream>>>(out, bias, nwsum, cnt, esc, n);
}